// CausalSelfAttention_60438779789934
// MI455X (gfx1250) — compile-verified
//
#include <hip/hip_runtime.h>
#include <math.h>

typedef float v2f __attribute__((ext_vector_type(2)));
typedef float v8f __attribute__((ext_vector_type(8)));

static __device__ __forceinline__ v8f wmma_f32(v2f a, v2f b, v8f c) {
  // V_WMMA_F32_16X16X4_F32 : D = A(16x4) * B(4x16) + C(16x16), full fp32
  return __builtin_amdgcn_wmma_f32_16x16x4_f32(false, a, false, b, (short)0, c,
                                               false, false);
}

// Low 32 bits of a generic pointer into __shared__ = LDS byte address
// (ISA 10.2: LDS aperture uses addr[31:0]).
static __device__ __forceinline__ unsigned lds_addr(const void* p) {
  return (unsigned)(unsigned long long)p;
}

// Direct memory->LDS DMA, tracked with ASYNCcnt (no VGPR staging).
static __device__ __forceinline__ void async_copy_b128(unsigned lds,
                                                       const void* gptr) {
  asm volatile("global_load_async_to_lds_b128 %0, %1, off" ::"v"(lds),
               "v"(gptr)
               : "memory");
}

static __device__ __forceinline__ void wait_async0() {
  asm volatile("s_wait_asynccnt 0x0" ::: "memory");
}

// ---------------------------------------------------------------------------
// GEMM (NT):  C[M,N] = A[M,K] * W[N,K]^T + bias[N]
// 128 threads (4 waves); 64x64 tile; K staged 32 at a time in LDS,
// double-buffered via async-to-LDS copies overlapped with WMMA.
// ---------------------------------------------------------------------------
#define GS 36  // padded LDS row stride (32+4): conflict-free, 16B aligned

__global__ __launch_bounds__(128) void gemm_nt_bias(
    const float* __restrict__ A, const float* __restrict__ W,
    const float* __restrict__ bias, float* __restrict__ C,
    int M, int N, int K) {
  __shared__ float As[2][64][GS];
  __shared__ float Bs[2][64][GS];

  const int t = threadIdx.x;
  const int wave = t >> 5, lane = t & 31;
  const int hi = lane >> 4, lo = lane & 15;
  const int koff = hi << 1;
  const int m0 = blockIdx.y * 64;
  const int n0 = blockIdx.x * 64;

  v8f acc[4];
#pragma unroll
  for (int i = 0; i < 4; ++i) acc[i] = (v8f){0, 0, 0, 0, 0, 0, 0, 0};

  // issue DMA for k-tile 0
#pragma unroll
  for (int i = 0; i < 4; ++i) {  // 64x32 floats = 512 float4, 4 per thread
    int lin = t + i * 128;
    int row = lin >> 3;
    int c4 = (lin & 7) << 2;
    async_copy_b128(lds_addr(&As[0][row][c4]),
                    &A[(size_t)(m0 + row) * K + c4]);
    async_copy_b128(lds_addr(&Bs[0][row][c4]),
                    &W[(size_t)(n0 + row) * K + c4]);
  }
  wait_async0();
  __syncthreads();

  const int nk = K >> 5;
  const int ar = wave * 16 + lo;
  for (int kt = 0; kt < nk; ++kt) {
    const int buf = kt & 1;
    if (kt + 1 < nk) {  // prefetch next k-tile while computing this one
      const int k0 = (kt + 1) << 5;
#pragma unroll
      for (int i = 0; i < 4; ++i) {
        int lin = t + i * 128;
        int row = lin >> 3;
        int c4 = (lin & 7) << 2;
        async_copy_b128(lds_addr(&As[buf ^ 1][row][c4]),
                        &A[(size_t)(m0 + row) * K + k0 + c4]);
        async_copy_b128(lds_addr(&Bs[buf ^ 1][row][c4]),
                        &W[(size_t)(n0 + row) * K + k0 + c4]);
      }
    }
#pragma unroll
    for (int kk = 0; kk < 32; kk += 4) {
      v2f a;
      a.x = As[buf][ar][kk + koff];
      a.y = As[buf][ar][kk + koff + 1];
#pragma unroll
      for (int nt = 0; nt < 4; ++nt) {
        v2f b;
        b.x = Bs[buf][nt * 16 + lo][kk + koff];
        b.y = Bs[buf][nt * 16 + lo][kk + koff + 1];
        acc[nt] = wmma_f32(a, b, acc[nt]);
      }
    }
    wait_async0();   // my DMA into buf^1 done
    __syncthreads(); // everyone done reading buf + their DMA done
  }

#pragma unroll
  for (int nt = 0; nt < 4; ++nt) {
    int col = n0 + nt * 16 + lo;
    float bv = bias[col];
#pragma unroll
    for (int r = 0; r < 8; ++r) {
      int row = m0 + wave * 16 + r + 8 * hi;
      C[(size_t)row * N + col] = acc[nt][r] + bv;
    }
  }
}

// ---------------------------------------------------------------------------
// Flash-style causal attention, one (b,h,64-row q-tile) per block.
// qkv layout: [B*L, 1920] with q|k|v sections of 640, head h at h*64.
// ---------------------------------------------------------------------------
#define AS 68  // padded LDS row stride (64+4)

__global__ __launch_bounds__(128) void attn_fwd(const float* __restrict__ qkv,
                                                float* __restrict__ y) {
  __shared__ float Qs[64][AS];
  __shared__ float KPs[64][AS];  // K tile, reused as P tile
  __shared__ float Vs[64][AS];

  const int t = threadIdx.x;
  const int wave = t >> 5, lane = t & 31;
  const int hi = lane >> 4, lo = lane & 15;
  const int koff = hi << 1;

  const int qt = blockIdx.x;  // 0..15
  const int bh = blockIdx.y;  // 0..79
  const int b = bh / 10, h = bh % 10;
  const int qrow0 = b * 1024 + qt * 64;  // global row base in [B*L]
  const int qcol = h * 64;
  const int kcolb = 640 + h * 64;
  const int vcolb = 1280 + h * 64;

#pragma unroll
  for (int i = 0; i < 8; ++i) {  // DMA Q tile (64x64) once
    int lin = t + i * 128;
    int row = lin >> 4;
    int c4 = (lin & 15) << 2;
    async_copy_b128(lds_addr(&Qs[row][c4]),
                    &qkv[(size_t)(qrow0 + row) * 1920 + qcol + c4]);
  }

  float m_s[8], l_s[8];
  v8f acc_o[4];
#pragma unroll
  for (int r = 0; r < 8; ++r) {
    m_s[r] = -INFINITY;
    l_s[r] = 0.f;
  }
#pragma unroll
  for (int i = 0; i < 4; ++i) acc_o[i] = (v8f){0, 0, 0, 0, 0, 0, 0, 0};

  const int ar = wave * 16 + lo;

  for (int jt = 0; jt <= qt; ++jt) {  // causal: only key tiles <= q tile
    int krow0 = b * 1024 + jt * 64;
#pragma unroll
    for (int i = 0; i < 8; ++i) {  // DMA K and V tiles
      int lin = t + i * 128;
      int row = lin >> 4;
      int c4 = (lin & 15) << 2;
      async_copy_b128(lds_addr(&KPs[row][c4]),
                      &qkv[(size_t)(krow0 + row) * 1920 + kcolb + c4]);
      async_copy_b128(lds_addr(&Vs[row][c4]),
                      &qkv[(size_t)(krow0 + row) * 1920 + vcolb + c4]);
    }
    wait_async0();
    __syncthreads();

    // --- S = Q * K^T  (64x64, K-dim = Dh = 64) ---
    v8f s[4];
#pragma unroll
    for (int i = 0; i < 4; ++i) s[i] = (v8f){0, 0, 0, 0, 0, 0, 0, 0};
#pragma unroll
    for (int kk = 0; kk < 64; kk += 4) {
      v2f a;
      a.x = Qs[ar][kk + koff];
      a.y = Qs[ar][kk + koff + 1];
#pragma unroll
      for (int nt = 0; nt < 4; ++nt) {
        v2f bf;
        bf.x = KPs[nt * 16 + lo][kk + koff];
        bf.y = KPs[nt * 16 + lo][kk + koff + 1];
        s[nt] = wmma_f32(a, bf, s[nt]);
      }
    }

    // --- scale + causal mask + row max ---
    float rowmax[8];
#pragma unroll
    for (int r = 0; r < 8; ++r) rowmax[r] = -INFINITY;
#pragma unroll
    for (int nt = 0; nt < 4; ++nt) {
      int kcol = jt * 64 + nt * 16 + lo;
#pragma unroll
      for (int r = 0; r < 8; ++r) {
        int qrow = qt * 64 + wave * 16 + r + 8 * hi;
        float v = s[nt][r] * 0.125f;  // Dh^-0.5
        v = (kcol <= qrow) ? v : -INFINITY;
        s[nt][r] = v;
        rowmax[r] = fmaxf(rowmax[r], v);
      }
    }
#pragma unroll
    for (int r = 0; r < 8; ++r)
#pragma unroll
      for (int off = 1; off < 16; off <<= 1)
        rowmax[r] = fmaxf(rowmax[r], __shfl_xor(rowmax[r], off, 32));

    // --- online softmax update ---
    float alpha[8], rowsum[8];
#pragma unroll
    for (int r = 0; r < 8; ++r) {
      float mnew = fmaxf(m_s[r], rowmax[r]);
      alpha[r] = __expf(m_s[r] - mnew);
      m_s[r] = mnew;
      rowsum[r] = 0.f;
    }
#pragma unroll
    for (int nt = 0; nt < 4; ++nt)
#pragma unroll
      for (int r = 0; r < 8; ++r) {
        float p = __expf(s[nt][r] - m_s[r]);
        s[nt][r] = p;
        rowsum[r] += p;
      }
#pragma unroll
    for (int r = 0; r < 8; ++r) {
#pragma unroll
      for (int off = 1; off < 16; off <<= 1)
        rowsum[r] += __shfl_xor(rowsum[r], off, 32);
      l_s[r] = l_s[r] * alpha[r] + rowsum[r];
    }
#pragma unroll
    for (int nt = 0; nt < 4; ++nt)
#pragma unroll
      for (int r = 0; r < 8; ++r) acc_o[nt][r] *= alpha[r];

    // --- D-layout -> A-layout for P via LDS (reuse K tile space) ---
    __syncthreads();  // all waves done reading KPs as K
#pragma unroll
    for (int nt = 0; nt < 4; ++nt)
#pragma unroll
      for (int r = 0; r < 8; ++r)
        KPs[wave * 16 + r + 8 * hi][nt * 16 + lo] = s[nt][r];
    __syncthreads();

    // --- O += P * V  (K-dim = 64 keys) ---
#pragma unroll
    for (int kk = 0; kk < 64; kk += 4) {
      v2f a;
      a.x = KPs[ar][kk + koff];
      a.y = KPs[ar][kk + koff + 1];
#pragma unroll
      for (int dt = 0; dt < 4; ++dt) {
        v2f bf;
        bf.x = Vs[kk + koff][dt * 16 + lo];
        bf.y = Vs[kk + koff + 1][dt * 16 + lo];
        acc_o[dt] = wmma_f32(a, bf, acc_o[dt]);
      }
    }
    __syncthreads();  // before next tile overwrites KPs/Vs
  }

  // --- normalize and write y[b,l, h*64 + d] ---
#pragma unroll
  for (int dt = 0; dt < 4; ++dt)
#pragma unroll
    for (int r = 0; r < 8; ++r) {
      int row = qrow0 + wave * 16 + r + 8 * hi;
      float o = acc_o[dt][r] / l_s[r];
      y[(size_t)row * 640 + h * 64 + dt * 16 + lo] = o;
    }
}

// ---------------------------------------------------------------------------
extern "C" void kernel_launch(void* const* d_in, const int* in_sizes, int n_in,
                              void* d_out, int out_size, void* d_ws,
                              size_t ws_size, hipStream_t stream) {
  const float* x = (const float*)d_in[0];      // [8,1024,640]
  const float* qkv_w = (const float*)d_in[1];  // [1920,640]
  const float* qkv_b = (const float*)d_in[2];  // [1920]
  const float* out_w = (const float*)d_in[3];  // [640,640]
  const float* out_b = (const float*)d_in[4];  // [640]
  float* out = (float*)d_out;                  // [8,1024,640]

  float* qkv = (float*)d_ws;             // [8192,1920]
  float* y = qkv + (size_t)8192 * 1920;  // [8192,640]

  dim3 blk(128);
  gemm_nt_bias<<<dim3(1920 / 64, 8192 / 64), blk, 0, stream>>>(
      x, qkv_w, qkv_b, qkv, 8192, 1920, 640);
  attn_fwd<<<dim3(16, 80), blk, 0, stream>>>(qkv, y);
  gemm_nt_bias<<<dim3(640 / 64, 8192 / 64), blk, 0, stream>>>(
      y, out_w, out_b, out, 8192, 640, 640);
}